// AdaptiveAvgPoolSequence_90537910600306
// MI455X (gfx1250) — compile-verified
//
#include <hip/hip_runtime.h>
#include <hip/hip_bf16.h>

// Problem constants (from reference)
#define BB   4
#define NN   65536
#define CC   256
#define HWB  64           // 8*8 bins
#define KK   16384        // HWB*CC
#define OUTC 512
#define KSPLIT 16
#define KCHUNK 256
#define APITCH 260        // KCHUNK + 4 to spread LDS banks

typedef __attribute__((ext_vector_type(2))) float v2f;
typedef __attribute__((ext_vector_type(8))) float v8f;

#if defined(__gfx1250__) && __has_builtin(__builtin_amdgcn_wmma_f32_16x16x4_f32)
#define HAVE_WMMA_F32X4 1
#else
#define HAVE_WMMA_F32X4 0
#endif

// ---------------------------------------------------------------------------
// Kernel 0: zero accumulators, seed output with bias
// ---------------------------------------------------------------------------
__global__ __launch_bounds__(256) void k_init(const float* __restrict__ bias,
                                              float* __restrict__ sums,
                                              float* __restrict__ counts,
                                              float* __restrict__ out) {
    int idx = blockIdx.x * 256 + threadIdx.x;          // 0..65535
    sums[idx] = 0.0f;
    if (idx < HWB) counts[idx] = 0.0f;
    if (idx < BB * OUTC) out[idx] = bias[idx & (OUTC - 1)];
}

// ---------------------------------------------------------------------------
// Kernel 1: bins from coords + per-bin counts (LDS histogram, one flush)
// ---------------------------------------------------------------------------
__global__ __launch_bounds__(256) void k_bins(const float* __restrict__ coords,
                                              int* __restrict__ bins,
                                              float* __restrict__ counts) {
    __shared__ int hist[HWB];
    int tid = threadIdx.x;
    if (tid < HWB) hist[tid] = 0;
    __syncthreads();

    int n = blockIdx.x * 256 + tid;
    float x = coords[2 * n + 0];
    float y = coords[2 * n + 1];
    const float lo   = -1.0f - 1e-6f;
    const float rstp = 8.0f / (2.0f + 2e-6f);          // 1/step
    int kx = (int)floorf((x - lo) * rstp);
    int ky = (int)floorf((y - lo) * rstp);
    kx = min(max(kx, 0), 7);
    ky = min(max(ky, 0), 7);
    int bin = kx + ky * 8;
    bins[n] = bin;
    atomicAdd(&hist[bin], 1);
    __syncthreads();
    if (tid < HWB && hist[tid] != 0) atomicAdd(&counts[tid], (float)hist[tid]);
}

// ---------------------------------------------------------------------------
// Kernel 2: segment sums. grid = (N/2048, B, 2 channel halves), block = 128.
// Each thread owns one channel column of all 64 bins in LDS -> no atomics in
// the 256MB streaming loop. values read once, non-temporal (dead data).
// ---------------------------------------------------------------------------
__global__ __launch_bounds__(128) void k_segsum(const float* __restrict__ values,
                                                const int* __restrict__ bins,
                                                float* __restrict__ sums) {
    __shared__ float acc[HWB * 128];   // 32 KB
    __shared__ int   sbin[2048];       //  8 KB
    int t  = threadIdx.x;              // 0..127
    int b  = blockIdx.y;               // batch
    int z  = blockIdx.z;               // channel half
    int n0 = blockIdx.x * 2048;
    int c  = z * 128 + t;

    #pragma unroll
    for (int i = 0; i < HWB; ++i) acc[i * 128 + t] = 0.0f;
    for (int i = t; i < 2048; i += 128) sbin[i] = bins[n0 + i];
    __syncthreads();

    const float* vp = values + ((size_t)b * NN + n0) * CC + c;
    #pragma unroll 4
    for (int n = 0; n < 2048; ++n) {
        int bin = sbin[n];
        acc[bin * 128 + t] += __builtin_nontemporal_load(vp + (size_t)n * CC);
    }
    __syncthreads();

    #pragma unroll
    for (int i = 0; i < HWB; ++i)
        atomicAdd(&sums[i * (BB * CC) + b * CC + c], acc[i * 128 + t]);
}

// ---------------------------------------------------------------------------
// Kernel 3: means -> zero-padded A matrix [16][16384] (rows 4..15 = 0)
// ---------------------------------------------------------------------------
__global__ __launch_bounds__(256) void k_finalize(const float* __restrict__ sums,
                                                  const float* __restrict__ counts,
                                                  float* __restrict__ Apad) {
    int idx = blockIdx.x * 256 + threadIdx.x;   // 0..262143
    int m   = idx >> 14;                        // 0..15
    int k   = idx & (KK - 1);
    int bin = k >> 8;
    int cc  = k & (CC - 1);
    float v = 0.0f;
    if (m < BB) {
        float cnt = counts[bin];
        v = sums[bin * (BB * CC) + m * CC + cc] / fmaxf(cnt, 1.0f);
    }
    Apad[(size_t)m * KK + k] = v;
}

// ---------------------------------------------------------------------------
// Kernel 4: GEMM [16(pad from 4) x 16384] x [16384 x 512] with
// V_WMMA_F32_16X16X4_F32. grid = KSPLIT blocks (each owns 1024 K values),
// 8 waves/block, each wave owns 4 of the 32 N-tiles. A chunk staged in LDS,
// W streamed (read exactly once grid-wide). f32 atomic accumulate into out.
// Fragment lane math per CDNA5 ISA 7.12.2 (wave32):
//   A 16x4:  vgpr0 = K=(0|2) for lanes (<16|>=16), vgpr1 = K+1, M = lane%16
//   B 4x16:  vgpr0 = row K=(0|2), vgpr1 = row K+1, N = lane%16
//   D 16x16: vgpr r = M=r (lanes 0-15) -> rows 0..3 are the real batch rows
// ---------------------------------------------------------------------------
__global__ __launch_bounds__(256) void k_gemm(const float* __restrict__ Apad,
                                              const float* __restrict__ W,
                                              float* __restrict__ out) {
    __shared__ float ldsA[16 * APITCH];   // ~16.6 KB
    int tid  = threadIdx.x;
    int wave = tid >> 5;
    int lane = tid & 31;
    int row  = lane & 15;
    int kb2  = (lane >> 4) << 1;          // 0 for lanes 0-15, 2 for 16-31
    int kbeg = blockIdx.x * (KK / KSPLIT);

    v8f acc[4] = {};

    for (int kc = 0; kc < KK / KSPLIT; kc += KCHUNK) {
        int k0 = kbeg + kc;
        __syncthreads();
        #pragma unroll
        for (int i = tid; i < 16 * KCHUNK; i += 256) {
            int r = i >> 8;               // i / KCHUNK
            int cidx = i & (KCHUNK - 1);
            ldsA[r * APITCH + cidx] = Apad[(size_t)r * KK + k0 + cidx];
            if (kc + KCHUNK < KK / KSPLIT)
                __builtin_prefetch(&Apad[(size_t)r * KK + k0 + KCHUNK + cidx], 0, 0);
        }
        __syncthreads();

        for (int s = 0; s < KCHUNK; s += 4) {
            const float* arow = &ldsA[row * APITCH + s + kb2];
            v2f a;
            a.x = arow[0];                // K = s+kb2
            a.y = arow[1];                // K = s+kb2+1
            #pragma unroll
            for (int i = 0; i < 4; ++i) {
                int jb = ((wave + (i << 3)) << 4) + row;
                const float* wp = &W[(size_t)(k0 + s + kb2) * OUTC + jb];
                v2f bf;
                bf.x = wp[0];             // row K = s+kb2
                bf.y = wp[OUTC];          // row K = s+kb2+1
#if HAVE_WMMA_F32X4
                acc[i] = __builtin_amdgcn_wmma_f32_16x16x4_f32(
                    false, a, false, bf, (short)0, acc[i], false, false);
#else
                // fallback (not expected): plain FMA into same fragment slots
                #pragma unroll
                for (int r4 = 0; r4 < 4; ++r4) {
                    float a0 = ldsA[r4 * APITCH + s + 0];
                    float a1 = ldsA[r4 * APITCH + s + 1];
                    float a2 = ldsA[r4 * APITCH + s + 2];
                    float a3 = ldsA[r4 * APITCH + s + 3];
                    const float* wq = &W[(size_t)(k0 + s) * OUTC + jb];
                    acc[i][r4] += a0 * wq[0] + a1 * wq[OUTC] +
                                  a2 * wq[2 * OUTC] + a3 * wq[3 * OUTC];
                }
#endif
            }
        }
    }

    // rows 0..3 (batch) live in acc vgprs 0..3, lanes 0..15
    if (lane < 16) {
        #pragma unroll
        for (int i = 0; i < 4; ++i) {
            int jb = ((wave + (i << 3)) << 4) + lane;
            #pragma unroll
            for (int r = 0; r < BB; ++r)
                atomicAdd(&out[r * OUTC + jb], acc[i][r]);
        }
    }
}

// ---------------------------------------------------------------------------
extern "C" void kernel_launch(void* const* d_in, const int* in_sizes, int n_in,
                              void* d_out, int out_size, void* d_ws, size_t ws_size,
                              hipStream_t stream) {
    const float* values = (const float*)d_in[0];   // [4, 65536, 256]
    const float* coords = (const float*)d_in[1];   // [65536, 2]
    const float* W      = (const float*)d_in[2];   // [16384, 512]
    const float* bias   = (const float*)d_in[3];   // [512]
    float* out = (float*)d_out;                    // [4, 512]

    float* ws     = (float*)d_ws;
    int*   bins   = (int*)ws;                      // 65536 ints
    float* counts = ws + NN;                       // 64
    float* sums   = counts + HWB;                  // 65536  ([bin][b][c])
    float* Apad   = sums + HWB * BB * CC;          // 262144 ([16][16384])

    k_init    <<<NN / 256, 256, 0, stream>>>(bias, sums, counts, out);
    k_bins    <<<NN / 256, 256, 0, stream>>>(coords, bins, counts);
    k_segsum  <<<dim3(NN / 2048, BB, 2), 128, 0, stream>>>(values, bins, sums);
    k_finalize<<<(16 * KK) / 256, 256, 0, stream>>>(sums, counts, Apad);
    k_gemm    <<<KSPLIT, 256, 0, stream>>>(Apad, W, out);
}